// Block_88845693485113
// MI455X (gfx1250) — compile-verified
//
#include <hip/hip_runtime.h>
#include <hip/hip_bf16.h>

// ViT block w/ DLA-refined attention for gfx1250 (MI455X).
// All dense GEMMs run on V_WMMA_F32_16X16X4_F32 (exact fp32 matrix math).

typedef float v2f __attribute__((ext_vector_type(2)));
typedef float v8f __attribute__((ext_vector_type(8)));

#define NTOK   197
#define NPIX   38809          // 197*197
#define BDIM   384
#define NB     32
#define NH     12
#define HIDCH  36
#define ROWS   6304           // 32*197
#define MLPH   1536

__device__ __forceinline__ float waveSum(float v) {
#pragma unroll
  for (int o = 16; o > 0; o >>= 1) v += __shfl_down(v, o, 32);
  return v;
}
__device__ __forceinline__ float waveMax(float v) {
#pragma unroll
  for (int o = 16; o > 0; o >>= 1) v = fmaxf(v, __shfl_down(v, o, 32));
  return v;
}
__device__ __forceinline__ float geluf(float v) {
  return 0.5f * v * (1.0f + erff(v * 0.70710678118654752f));
}

// ---------------------------------------------------------------------------
// Batched GEMM:  C[m,n] = sum_k A[m,k] * W[n,k]   (i.e. C = A @ W^T)
// 256 threads = 8 waves; block tile 128x64, K staged 32 deep, double-buffered
// LDS with register staging (1 barrier / K-tile). LDS holds K-pairs as float2
// so each WMMA fragment is a single ds_load_b64; k2-row stride is padded so
// the two 16-lane halves hit disjoint bank sets.
// Each wave owns a 32x32 tile: 4 accumulators, 4 WMMAs per 4 fragment loads.
// epMode: 0 = none, 1 = +bias[n], 2 = +bias[n] then exact GELU.
// ---------------------------------------------------------------------------
#define BM 128
#define BN 64
#define BK 32
#define ASTR (2 * BM + 32)    // 288 floats per k2 row (dword stride % 64 == 32)
#define WSTR (2 * BN + 32)    // 160 floats per k2 row

__global__ __launch_bounds__(256) void gemm_wmma_f32(
    const float* __restrict__ A, long aBatch, int lda,
    const float* __restrict__ W, long wBatch, int ldw,
    float* __restrict__ C, long cBatch, int ldc,
    int M, int N, int K, int epMode, const float* __restrict__ bias)
{
  __shared__ float As[2][16 * ASTR];
  __shared__ float Ws[2][16 * WSTR];
  const float* Ab = A + (long)blockIdx.z * aBatch;
  const float* Wb = W + (long)blockIdx.z * wBatch;
  float*       Cb = C + (long)blockIdx.z * cBatch;
  const int m0  = blockIdx.y * BM;
  const int n0  = blockIdx.x * BN;
  const int tid = threadIdx.x;
  const int lane = tid & 31;
  const int half = lane >> 4;      // 0: lanes 0-15, 1: lanes 16-31
  const int l16  = lane & 15;
  const int wave = tid >> 5;
  const int mw = (wave >> 1) * 32; // wave row offset: 0/32/64/96
  const int nw = (wave & 1) * 32;  // wave col offset: 0/32

  v8f acc00 = {0.f,0.f,0.f,0.f,0.f,0.f,0.f,0.f};
  v8f acc01 = acc00, acc10 = acc00, acc11 = acc00;

  // staging thread mapping
  const int am = tid >> 1;          // 0..127  (A row)
  const int ak = (tid & 1) << 4;    // 0,16    (A k segment, 16 wide)
  const int wn = tid >> 2;          // 0..63   (W row)
  const int wk = (tid & 3) << 3;    // 0,8,16,24 (W k segment, 8 wide)

  float ra[16];
  float rw[8];
  const int kTiles = (K + BK - 1) / BK;

  auto loadTile = [&](int kt) {
    const int k0 = kt * BK;
    const int  gm  = m0 + am;
    const bool mOK = gm < M;
    const float* src = Ab + (long)gm * lda + k0 + ak;
#pragma unroll
    for (int i = 0; i < 16; ++i) {
      const int gk = k0 + ak + i;
      ra[i] = (mOK && gk < K) ? src[i] : 0.0f;
    }
    const int  gn  = n0 + wn;
    const bool nOK = gn < N;
    const float* wsrc = Wb + (long)gn * ldw + k0 + wk;
#pragma unroll
    for (int i = 0; i < 8; ++i) {
      const int gk = k0 + wk + i;
      rw[i] = (nOK && gk < K) ? wsrc[i] : 0.0f;
    }
    if (kt + 1 < kTiles) {
      if (mOK) __builtin_prefetch(src + BK, 0, 1);
      if (nOK) __builtin_prefetch(wsrc + BK, 0, 1);
    }
  };
  auto storeTile = [&](int buf) {
#pragma unroll
    for (int j = 0; j < 8; ++j) {
      const int k2 = (ak >> 1) + j;
      As[buf][k2 * ASTR + 2 * am + 0] = ra[2 * j + 0];
      As[buf][k2 * ASTR + 2 * am + 1] = ra[2 * j + 1];
    }
#pragma unroll
    for (int j = 0; j < 4; ++j) {
      const int k2 = (wk >> 1) + j;
      Ws[buf][k2 * WSTR + 2 * wn + 0] = rw[2 * j + 0];
      Ws[buf][k2 * WSTR + 2 * wn + 1] = rw[2 * j + 1];
    }
  };

  loadTile(0);
  storeTile(0);
  int cur = 0;
  for (int kt = 0; kt < kTiles; ++kt) {
    if (kt + 1 < kTiles) loadTile(kt + 1);
    __syncthreads();   // buf[cur] writes visible; buf[cur^1] readers retired
    const float* Ac = As[cur];
    const float* Wc = Ws[cur];
#pragma unroll
    for (int kk = 0; kk < BK; kk += 4) {
      const int k2 = (kk >> 1) + half;
      // A 16x4 frag: lanes 0-15 hold K=kk+0/1, lanes 16-31 hold K=kk+2/3
      v2f a0 = *(const v2f*)&Ac[k2 * ASTR + 2 * (mw + l16)];
      v2f a1 = *(const v2f*)&Ac[k2 * ASTR + 2 * (mw + 16 + l16)];
      v2f b0 = *(const v2f*)&Wc[k2 * WSTR + 2 * (nw + l16)];
      v2f b1 = *(const v2f*)&Wc[k2 * WSTR + 2 * (nw + 16 + l16)];
      acc00 = __builtin_amdgcn_wmma_f32_16x16x4_f32(false, a0, false, b0,
                                                    (short)0, acc00, false, false);
      acc01 = __builtin_amdgcn_wmma_f32_16x16x4_f32(false, a0, false, b1,
                                                    (short)0, acc01, false, false);
      acc10 = __builtin_amdgcn_wmma_f32_16x16x4_f32(false, a1, false, b0,
                                                    (short)0, acc10, false, false);
      acc11 = __builtin_amdgcn_wmma_f32_16x16x4_f32(false, a1, false, b1,
                                                    (short)0, acc11, false, false);
    }
    if (kt + 1 < kTiles) storeTile(cur ^ 1);
    cur ^= 1;
  }

  const int gnA = n0 + nw + l16;
  const int gnB = gnA + 16;
  float bA = 0.0f, bB = 0.0f;
  if (epMode >= 1) {
    if (gnA < N) bA = bias[gnA];
    if (gnB < N) bB = bias[gnB];
  }
#pragma unroll
  for (int i = 0; i < 8; ++i) {
    // C/D layout: VGPR i -> row i (lanes 0-15) / row 8+i (lanes 16-31)
    const int r0 = m0 + mw + (half << 3) + i;
    const int r1 = r0 + 16;
    if (gnA < N) {
      if (r0 < M) {
        float v = acc00[i] + bA;
        if (epMode == 2) v = geluf(v);
        Cb[(long)r0 * ldc + gnA] = v;
      }
      if (r1 < M) {
        float v = acc10[i] + bA;
        if (epMode == 2) v = geluf(v);
        Cb[(long)r1 * ldc + gnA] = v;
      }
    }
    if (gnB < N) {
      if (r0 < M) {
        float v = acc01[i] + bB;
        if (epMode == 2) v = geluf(v);
        Cb[(long)r0 * ldc + gnB] = v;
      }
      if (r1 < M) {
        float v = acc11[i] + bB;
        if (epMode == 2) v = geluf(v);
        Cb[(long)r1 * ldc + gnB] = v;
      }
    }
  }
}

// ---------------------------------------------------------------------------
// LayerNorm over last dim (cols=384) of premul*x.
// ---------------------------------------------------------------------------
__global__ __launch_bounds__(256) void ln_kernel(
    const float* __restrict__ x, const float* __restrict__ g,
    const float* __restrict__ b, float* __restrict__ out,
    int cols, float premul)
{
  const long row = blockIdx.x;
  const float* xr = x + row * cols;
  float* yr = out + row * cols;
  const int t = threadIdx.x;
  float s = 0.f, q = 0.f;
  for (int c = t; c < cols; c += 256) {
    float v = xr[c] * premul; s += v; q += v * v;
  }
  __shared__ float r1[8], r2[8];
  s = waveSum(s); q = waveSum(q);
  if ((t & 31) == 0) { r1[t >> 5] = s; r2[t >> 5] = q; }
  __syncthreads();
  float S = 0.f, Q = 0.f;
  for (int i = 0; i < 8; ++i) { S += r1[i]; Q += r2[i]; }
  const float mean = S / cols;
  const float var  = Q / cols - mean * mean;
  const float inv  = rsqrtf(var + 1e-5f);
  for (int c = t; c < cols; c += 256) {
    float v = xr[c] * premul;
    yr[c] = (v - mean) * inv * g[c] + b[c];
  }
}

// ---------------------------------------------------------------------------
// Row softmax, cols = 197, in place.
// ---------------------------------------------------------------------------
__global__ __launch_bounds__(256) void softmax_kernel(float* __restrict__ S, int cols)
{
  const long row = blockIdx.x;
  float* r = S + row * cols;
  const int t = threadIdx.x;
  float v = (t < cols) ? r[t] : -3.0e38f;
  __shared__ float rm[8], rs[8];
  float m = waveMax(v);
  if ((t & 31) == 0) rm[t >> 5] = m;
  __syncthreads();
  float M = rm[0];
  for (int i = 1; i < 8; ++i) M = fmaxf(M, rm[i]);
  float e = (t < cols) ? expf(v - M) : 0.0f;
  float s = waveSum(e);
  if ((t & 31) == 0) rs[t >> 5] = s;
  __syncthreads();
  float T = 0.f;
  for (int i = 0; i < 8; ++i) T += rs[i];
  if (t < cols) r[t] = e / T;
}

// ---------------------------------------------------------------------------
// Scatter qkv [6304,1152] -> q/k scaled [B,H,N,32], vT [B,H,32,N].
// ---------------------------------------------------------------------------
__global__ void repack_qkv(const float* __restrict__ qkv, float* __restrict__ q,
                           float* __restrict__ k, float* __restrict__ vT, float sq)
{
  const long idx = (long)blockIdx.x * 256 + threadIdx.x;
  if (idx >= (long)ROWS * 1152) return;
  const int  c   = (int)(idx % 1152);
  const long row = idx / 1152;
  const int  n = (int)(row % NTOK);
  const int  b = (int)(row / NTOK);
  const int  tsel = c / BDIM;
  const int  hc = c % BDIM;
  const int  h = hc >> 5, d = hc & 31;
  const float v = qkv[idx];
  const long bh = (long)b * NH + h;
  if (tsel == 0)      q[(bh * NTOK + n) * 32 + d] = v * sq;
  else if (tsel == 1) k[(bh * NTOK + n) * 32 + d] = v * sq;
  else                vT[(bh * 32 + d) * NTOK + n] = v;
}

// ---------------------------------------------------------------------------
// DLA 1x1 expand (12->36) + BN1 batch-stat accumulation; stores ae.
// ---------------------------------------------------------------------------
__global__ __launch_bounds__(256) void dla_expand_stats(
    const float* __restrict__ a0, const float* __restrict__ ew,
    float* __restrict__ ae, float* __restrict__ sums)
{
  const int b = blockIdx.y;
  const int p = blockIdx.x * 256 + threadIdx.x;
  const bool ok = p < NPIX;
  __shared__ float sew[HIDCH * NH];
  __shared__ float ssum[HIDCH], ssq[HIDCH];
  for (int i = threadIdx.x; i < HIDCH * NH; i += 256) sew[i] = ew[i];
  if (threadIdx.x < HIDCH) { ssum[threadIdx.x] = 0.f; ssq[threadIdx.x] = 0.f; }
  __syncthreads();
  float in[NH];
  const long pixBase = (long)b * NH * NPIX + p;
#pragma unroll
  for (int c = 0; c < NH; ++c) in[c] = ok ? a0[pixBase + (long)c * NPIX] : 0.0f;
  const long oBase = (long)b * HIDCH * NPIX + p;
  const int lane = threadIdx.x & 31;
  for (int o = 0; o < HIDCH; ++o) {
    float v = 0.f;
#pragma unroll
    for (int c = 0; c < NH; ++c) v += in[c] * sew[o * NH + c];
    if (ok) ae[oBase + (long)o * NPIX] = v;
    float s = waveSum(v), q = waveSum(v * v);
    if (lane == 0) { atomicAdd(&ssum[o], s); atomicAdd(&ssq[o], q); }
  }
  __syncthreads();
  if (threadIdx.x < HIDCH) {
    atomicAdd(&sums[threadIdx.x], ssum[threadIdx.x]);
    atomicAdd(&sums[HIDCH + threadIdx.x], ssq[threadIdx.x]);
  }
}

// sums -> per-channel affine (scale, shift) for BN.
__global__ void bn_finalize(const float* __restrict__ sums, const float* __restrict__ g,
                            const float* __restrict__ b, float* __restrict__ ss,
                            int C, float invCnt)
{
  const int c = threadIdx.x;
  if (c >= C) return;
  const float mean = sums[c] * invCnt;
  const float var  = sums[C + c] * invCnt - mean * mean;
  const float inv  = rsqrtf(var + 1e-5f);
  ss[c]     = g[c] * inv;
  ss[C + c] = b[c] - mean * g[c] * inv;
}

__global__ void bn_apply_relu6(float* __restrict__ a, const float* __restrict__ ss,
                               int nch, long total)
{
  const long idx = (long)blockIdx.x * 256 + threadIdx.x;
  if (idx >= total) return;
  const int c = (int)((idx / NPIX) % nch);
  float v = a[idx] * ss[c] + ss[nch + c];
  a[idx] = fminf(fmaxf(v, 0.f), 6.f);
}

// ---------------------------------------------------------------------------
// Depthwise 3x3 (SAME) computed on the fly; accumulate BN2 stats only.
// ---------------------------------------------------------------------------
__global__ __launch_bounds__(256) void dw_stats(
    const float* __restrict__ a1, const float* __restrict__ dww,
    float* __restrict__ sums)
{
  const int o = blockIdx.y, b = blockIdx.z;
  const int p = blockIdx.x * 256 + threadIdx.x;
  const bool ok = p < NPIX;
  float w[9];
#pragma unroll
  for (int t = 0; t < 9; ++t) w[t] = dww[o * 9 + t];
  float v = 0.f;
  if (ok) {
    const float* plane = a1 + ((long)b * HIDCH + o) * NPIX;
    const int i = p / NTOK, j = p - i * NTOK;
#pragma unroll
    for (int di = -1; di <= 1; ++di) {
      const int ii = i + di;
      if (ii < 0 || ii >= NTOK) continue;
#pragma unroll
      for (int dj = -1; dj <= 1; ++dj) {
        const int jj = j + dj;
        if (jj < 0 || jj >= NTOK) continue;
        v += plane[ii * NTOK + jj] * w[(di + 1) * 3 + (dj + 1)];
      }
    }
  }
  __shared__ float r1[8], r2[8];
  float s = waveSum(v), q = waveSum(v * v);
  const int lane = threadIdx.x & 31, wv = threadIdx.x >> 5;
  if (lane == 0) { r1[wv] = s; r2[wv] = q; }
  __syncthreads();
  if (threadIdx.x == 0) {
    float S = 0.f, Q = 0.f;
    for (int i2 = 0; i2 < 8; ++i2) { S += r1[i2]; Q += r2[i2]; }
    atomicAdd(&sums[o], S);
    atomicAdd(&sums[HIDCH + o], Q);
  }
}

// ---------------------------------------------------------------------------
// Recompute dw3x3 -> BN2+ReLU6 -> 1x1 project (36->12); store a3 + BN3 stats.
// ---------------------------------------------------------------------------
__global__ __launch_bounds__(256) void dla_proj_stats(
    const float* __restrict__ a1, const float* __restrict__ ss2,
    const float* __restrict__ dww, const float* __restrict__ pw,
    float* __restrict__ a3, float* __restrict__ sums)
{
  const int b = blockIdx.y;
  const int p = blockIdx.x * 256 + threadIdx.x;
  const bool ok = p < NPIX;
  __shared__ float spw[NH * HIDCH];
  __shared__ float sdw[HIDCH * 9];
  __shared__ float sss[2 * HIDCH];
  __shared__ float ssum[NH], ssq[NH];
  for (int i = threadIdx.x; i < NH * HIDCH; i += 256) spw[i] = pw[i];
  for (int i = threadIdx.x; i < HIDCH * 9; i += 256) sdw[i] = dww[i];
  for (int i = threadIdx.x; i < 2 * HIDCH; i += 256) sss[i] = ss2[i];
  if (threadIdx.x < NH) { ssum[threadIdx.x] = 0.f; ssq[threadIdx.x] = 0.f; }
  __syncthreads();
  const int i0 = ok ? p / NTOK : 0;
  const int j0 = ok ? p - i0 * NTOK : 0;
  float t[HIDCH];
  for (int o = 0; o < HIDCH; ++o) {
    float v = 0.f;
    if (ok) {
      const float* plane = a1 + ((long)b * HIDCH + o) * NPIX;
      for (int di = -1; di <= 1; ++di) {
        const int ii = i0 + di;
        if (ii < 0 || ii >= NTOK) continue;
        for (int dj = -1; dj <= 1; ++dj) {
          const int jj = j0 + dj;
          if (jj < 0 || jj >= NTOK) continue;
          v += plane[ii * NTOK + jj] * sdw[o * 9 + (di + 1) * 3 + (dj + 1)];
        }
      }
      v = v * sss[o] + sss[HIDCH + o];
      v = fminf(fmaxf(v, 0.f), 6.f);
    }
    t[o] = v;
  }
  const int lane = threadIdx.x & 31;
  for (int h = 0; h < NH; ++h) {
    float v = 0.f;
#pragma unroll
    for (int o = 0; o < HIDCH; ++o) v += t[o] * spw[h * HIDCH + o];
    if (!ok) v = 0.f;
    if (ok) a3[((long)b * NH + h) * NPIX + p] = v;
    float s = waveSum(v), q = waveSum(v * v);
    if (lane == 0) { atomicAdd(&ssum[h], s); atomicAdd(&ssq[h], q); }
  }
  __syncthreads();
  if (threadIdx.x < NH) {
    atomicAdd(&sums[threadIdx.x], ssum[threadIdx.x]);
    atomicAdd(&sums[NH + threadIdx.x], ssq[threadIdx.x]);
  }
}

// t = bn3(a3) + a0 (in place on a3) + accumulate adapt-BN stats.
__global__ __launch_bounds__(256) void bn_resid_stats(
    float* __restrict__ a3, const float* __restrict__ a0,
    const float* __restrict__ ss, float* __restrict__ sums, long total)
{
  __shared__ float ssum[NH], ssq[NH];
  const int t = threadIdx.x;
  if (t < NH) { ssum[t] = 0.f; ssq[t] = 0.f; }
  __syncthreads();
  const long idx = (long)blockIdx.x * 256 + t;
  if (idx < total) {
    const int c = (int)((idx / NPIX) % NH);
    float v = a3[idx] * ss[c] + ss[NH + c] + a0[idx];
    a3[idx] = v;
    atomicAdd(&ssum[c], v);
    atomicAdd(&ssq[c], v * v);
  }
  __syncthreads();
  if (t < NH) {
    atomicAdd(&sums[t], ssum[t]);
    atomicAdd(&sums[NH + t], ssq[t]);
  }
}

__global__ void bn_apply(const float* __restrict__ a, const float* __restrict__ ss,
                         int nch, float* __restrict__ out, long total)
{
  const long idx = (long)blockIdx.x * 256 + threadIdx.x;
  if (idx >= total) return;
  const int c = (int)((idx / NPIX) % nch);
  out[idx] = a[idx] * ss[c] + ss[nch + c];
}

// [bh,197,32] -> [6304,384]
__global__ void merge_heads(const float* __restrict__ av, float* __restrict__ out)
{
  const long idx = (long)blockIdx.x * 256 + threadIdx.x;
  if (idx >= (long)ROWS * BDIM) return;
  const int  c   = (int)(idx % BDIM);
  const long row = idx / BDIM;
  const int n = (int)(row % NTOK);
  const int b = (int)(row / NTOK);
  const int h = c >> 5, d = c & 31;
  out[idx] = av[(((long)b * NH + h) * NTOK + n) * 32 + d];
}

__global__ void resid_add2(const float* __restrict__ x, const float* __restrict__ p,
                           float* __restrict__ y, long total)
{
  const long idx = (long)blockIdx.x * 256 + threadIdx.x;
  if (idx >= total) return;
  y[idx] = x[idx] + 2.0f * p[idx];
}

__global__ void final_resid(float* __restrict__ xio, const float* __restrict__ m2,
                            const float* __restrict__ sc, long total)
{
  const long idx = (long)blockIdx.x * 256 + threadIdx.x;
  if (idx >= total) return;
  const int c = (int)(idx % BDIM);
  xio[idx] += 2.0f * m2[idx] * sc[c];
}

// ---------------------------------------------------------------------------
extern "C" void kernel_launch(void* const* d_in, const int* in_sizes, int n_in,
                              void* d_out, int out_size, void* d_ws, size_t ws_size,
                              hipStream_t stream)
{
  const float* x      = (const float*)d_in[0];
  const float* ln1_g  = (const float*)d_in[1];
  const float* ln1_b  = (const float*)d_in[2];
  const float* qkv_w  = (const float*)d_in[3];
  const float* exp_w  = (const float*)d_in[4];
  const float* bn1_g  = (const float*)d_in[5];
  const float* bn1_b  = (const float*)d_in[6];
  const float* dw_w   = (const float*)d_in[7];
  const float* bn2_g  = (const float*)d_in[8];
  const float* bn2_b  = (const float*)d_in[9];
  const float* pro_w  = (const float*)d_in[10];
  const float* bn3_g  = (const float*)d_in[11];
  const float* bn3_b  = (const float*)d_in[12];
  const float* abn_g  = (const float*)d_in[13];
  const float* abn_b  = (const float*)d_in[14];
  const float* proj_w = (const float*)d_in[15];
  const float* proj_b = (const float*)d_in[16];
  const float* ln2_g  = (const float*)d_in[17];
  const float* ln2_b  = (const float*)d_in[18];
  const float* fc1_w  = (const float*)d_in[19];
  const float* fc1_b  = (const float*)d_in[20];
  const float* fc2_w  = (const float*)d_in[21];
  const float* fc2_b  = (const float*)d_in[22];
  const float* sc_ch  = (const float*)d_in[23];

  const long NXD  = (long)ROWS * BDIM;        // 2,420,736
  const long NQKV = (long)ROWS * 3 * BDIM;    // 7,262,208
  const long NATT = (long)NB * NH * NPIX;     // 14,902,656
  const long NAE  = (long)NB * HIDCH * NPIX;  // 44,707,968

  float* ws   = (float*)d_ws;
  float* h    = ws;                 // 6304x384
  float* qkv  = h + NXD;            // 6304x1152 (reused later)
  float* q    = qkv + NQKV;         // [B,H,N,32]
  float* k    = q + NXD;
  float* vT   = k + NXD;            // [B,H,32,N]
  float* a0   = vT + NXD;           // scores -> softmaxed attn (pre-DLA)
  float* ae   = a0 + NATT;          // [B,36,N,N] expanded map (reused as m1)
  float* a3   = ae + NAE;           // [B,12,N,N]
  float* st   = a3 + NATT;          // 384 stats floats
  // aliases (qkv dead after repack; ae dead after dla_proj_stats)
  float* avtmp = qkv;               // [bh,197,32]
  float* outb  = qkv + NXD;         // [6304,384]
  float* pbuf  = qkv + 2 * NXD;     // [6304,384]
  float* m1    = ae;                // [6304,1536]

  float* sums1 = st;       float* ss1 = st + 72;
  float* sums2 = st + 144; float* ss2 = st + 216;
  float* sums3 = st + 288; float* ss3 = st + 312;
  float* sums4 = st + 336; float* ss4 = st + 360;

  float* xout     = (float*)d_out;
  float* attn_out = xout + NXD;

  const float sq = 0.4204482076268573f;        // 32^-0.25
  const float invCnt = 1.0f / 1241888.0f;      // 1 / (32*197*197)

  hipMemsetAsync(st, 0, 384 * sizeof(float), stream);

  // LN1 on 0.5*x
  ln_kernel<<<ROWS, 256, 0, stream>>>(x, ln1_g, ln1_b, h, BDIM, 0.5f);
  // QKV GEMM
  gemm_wmma_f32<<<dim3(18, 50, 1), 256, 0, stream>>>(
      h, 0, BDIM, qkv_w, 0, BDIM, qkv, 0, 3 * BDIM, ROWS, 3 * BDIM, BDIM, 0, nullptr);
  repack_qkv<<<(NQKV + 255) / 256, 256, 0, stream>>>(qkv, q, k, vT, sq);
  // scores = q @ k^T (batched over 384 bh)
  gemm_wmma_f32<<<dim3(4, 2, NB * NH), 256, 0, stream>>>(
      q, (long)NTOK * 32, 32, k, (long)NTOK * 32, 32,
      a0, (long)NPIX, NTOK, NTOK, NTOK, 32, 0, nullptr);
  softmax_kernel<<<(long)NB * NH * NTOK, 256, 0, stream>>>(a0, NTOK);

  // DLA refinement
  dla_expand_stats<<<dim3(152, NB), 256, 0, stream>>>(a0, exp_w, ae, sums1);
  bn_finalize<<<1, 64, 0, stream>>>(sums1, bn1_g, bn1_b, ss1, HIDCH, invCnt);
  bn_apply_relu6<<<(NAE + 255) / 256, 256, 0, stream>>>(ae, ss1, HIDCH, NAE);
  dw_stats<<<dim3(152, HIDCH, NB), 256, 0, stream>>>(ae, dw_w, sums2);
  bn_finalize<<<1, 64, 0, stream>>>(sums2, bn2_g, bn2_b, ss2, HIDCH, invCnt);
  dla_proj_stats<<<dim3(152, NB), 256, 0, stream>>>(ae, ss2, dw_w, pro_w, a3, sums3);
  bn_finalize<<<1, 64, 0, stream>>>(sums3, bn3_g, bn3_b, ss3, NH, invCnt);
  bn_resid_stats<<<(NATT + 255) / 256, 256, 0, stream>>>(a3, a0, ss3, sums4, NATT);
  bn_finalize<<<1, 64, 0, stream>>>(sums4, abn_g, abn_b, ss4, NH, invCnt);
  bn_apply<<<(NATT + 255) / 256, 256, 0, stream>>>(a3, ss4, NH, attn_out, NATT);

  // out = attn @ v  (vT is [bh,32,197] so this is A @ W^T form)
  gemm_wmma_f32<<<dim3(1, 2, NB * NH), 256, 0, stream>>>(
      attn_out, (long)NPIX, NTOK, vT, (long)32 * NTOK, NTOK,
      avtmp, (long)NTOK * 32, 32, NTOK, 32, NTOK, 0, nullptr);
  merge_heads<<<(NXD + 255) / 256, 256, 0, stream>>>(avtmp, outb);
  // proj + bias, then x_mid = x + 2*proj_out
  gemm_wmma_f32<<<dim3(6, 50, 1), 256, 0, stream>>>(
      outb, 0, BDIM, proj_w, 0, BDIM, pbuf, 0, BDIM, ROWS, BDIM, BDIM, 1, proj_b);
  resid_add2<<<(NXD + 255) / 256, 256, 0, stream>>>(x, pbuf, xout, NXD);

  // MLP
  ln_kernel<<<ROWS, 256, 0, stream>>>(xout, ln2_g, ln2_b, h, BDIM, 0.5f);
  gemm_wmma_f32<<<dim3(24, 50, 1), 256, 0, stream>>>(
      h, 0, BDIM, fc1_w, 0, BDIM, m1, 0, MLPH, ROWS, MLPH, BDIM, 2, fc1_b);
  gemm_wmma_f32<<<dim3(6, 50, 1), 256, 0, stream>>>(
      m1, 0, MLPH, fc2_w, 0, MLPH, pbuf, 0, BDIM, ROWS, BDIM, MLPH, 1, fc2_b);
  final_resid<<<(NXD + 255) / 256, 256, 0, stream>>>(xout, pbuf, sc_ch, NXD);
}